// OptimizedSpatialInSARModel_85779086835980
// MI455X (gfx1250) — compile-verified
//
#include <hip/hip_runtime.h>
#include <math.h>

typedef __attribute__((ext_vector_type(2))) float v2f;
typedef __attribute__((ext_vector_type(8))) float v8f;

#define N_STATIONS 100000
#define N_TIME     1000
#define K_NEIGH    8
#define SMOOTH     0.2f
#define NCOEF      12            // K dim padded to multiple of 4 (10 -> 12)
#define BP         1008          // time dim padded to 63 * 16
#define M_TILES    (N_STATIONS / 16)   // 6250 (exact)
#define N_TILES    (BP / 16)           // 63
#define TWO_PI_F   6.28318530717958647692f

// ---------------------------------------------------------------------------
// Phase A: per-station coefficients  S[n][0..11]
//   S[0]=const, S[1]=trend, S[2+c]=amp_s*cos(ph_s), S[6+c]=amp_s*sin(ph_s)
//   (atan2 eliminated: cos/sin of atan2(im,re) = re,im * rsqrt(re^2+im^2))
// ---------------------------------------------------------------------------
__global__ __launch_bounds__(256) void coeff_kernel(
    const float* __restrict__ amps,     // [N,4]
    const float* __restrict__ phases,   // [N,4]
    const int*   __restrict__ nbr_idx,  // [N,8]
    const float* __restrict__ nbr_w,    // [N,8]
    const float* __restrict__ coff,     // [N]
    const float* __restrict__ trend,    // [N]
    float*       __restrict__ S)        // [N,12]
{
    int n = blockIdx.x * blockDim.x + threadIdx.x;
    if (n >= N_STATIONS) return;

    const float4* amps4   = (const float4*)amps;
    const float4* phases4 = (const float4*)phases;

    float wa[4] = {0.f, 0.f, 0.f, 0.f};
    float re[4] = {0.f, 0.f, 0.f, 0.f};
    float im[4] = {0.f, 0.f, 0.f, 0.f};

    #pragma unroll
    for (int k = 0; k < K_NEIGH; ++k) {
        int   j = nbr_idx[n * K_NEIGH + k];
        float w = nbr_w  [n * K_NEIGH + k];
        float4 a = amps4[j];
        float4 p = phases4[j];
        float av[4] = {a.x, a.y, a.z, a.w};
        float pv[4] = {p.x, p.y, p.z, p.w};
        #pragma unroll
        for (int c = 0; c < 4; ++c) {
            wa[c] = fmaf(w, av[c], wa[c]);
            float s, co;
            __sincosf(pv[c], &s, &co);
            re[c] = fmaf(w, co, re[c]);
            im[c] = fmaf(w, s,  im[c]);
        }
    }

    float4 a_own = amps4[n];
    float4 p_own = phases4[n];
    float aov[4] = {a_own.x, a_own.y, a_own.z, a_own.w};
    float pov[4] = {p_own.x, p_own.y, p_own.z, p_own.w};

    float* Srow = S + (size_t)n * NCOEF;
    Srow[0] = coff[n];
    Srow[1] = trend[n];
    #pragma unroll
    for (int c = 0; c < 4; ++c) {
        float amp_s = (1.0f - SMOOTH) * aov[c] + SMOOTH * wa[c];
        float s0, c0;
        __sincosf(pov[c], &s0, &c0);
        float mre = (1.0f - SMOOTH) * c0 + SMOOTH * re[c];
        float mim = (1.0f - SMOOTH) * s0 + SMOOTH * im[c];
        float rr  = fmaf(mre, mre, mim * mim);
        float inv = rsqrtf(fmaxf(rr, 1e-30f));
        Srow[2 + c] = amp_s * mre * inv;   // pairs with sin(w t)
        Srow[6 + c] = amp_s * mim * inv;   // pairs with cos(w t)
    }
    Srow[10] = 0.0f;
    Srow[11] = 0.0f;
}

// ---------------------------------------------------------------------------
// Phase A2: time-basis matrix  Bmat[12][1008]
// ---------------------------------------------------------------------------
__global__ __launch_bounds__(256) void basis_kernel(
    const float* __restrict__ tv,       // [1000]
    const float* __restrict__ periods,  // [4]
    float*       __restrict__ Bmat)     // [12,1008]
{
    int t = blockIdx.x * blockDim.x + threadIdx.x;
    if (t >= BP) return;
    float tt = (t < N_TIME) ? tv[t] : 0.0f;
    Bmat[0 * BP + t] = 1.0f;
    Bmat[1 * BP + t] = tt;
    #pragma unroll
    for (int i = 0; i < 4; ++i) {
        float w = TWO_PI_F / periods[i];
        float s, c;
        __sincosf(w * tt, &s, &c);
        Bmat[(2 + i) * BP + t] = s;
        Bmat[(6 + i) * BP + t] = c;
    }
    Bmat[10 * BP + t] = 0.0f;
    Bmat[11 * BP + t] = 0.0f;
}

// ---------------------------------------------------------------------------
// Phase B: out[100000,1000] = S[100000,12] x Bmat[12,1008] via WMMA f32 16x16x4
//   wave -> one 16-row M tile, loops over 8 N tiles reusing the A fragments.
//   A frag layout: lanes 0-15 hold M=lane, K={0,1}; lanes 16-31 K={2,3}.
//   B frag layout: lanes 0-15 hold N=lane, K={0,1}; lanes 16-31 K={2,3}.
//   C/D: VGPR r holds row r (lanes 0-15) / row r+8 (lanes 16-31).
// ---------------------------------------------------------------------------
__global__ __launch_bounds__(256) void gemm_kernel(
    const float* __restrict__ S,        // [100000,12]
    const float* __restrict__ Bmat,     // [12,1008]
    float*       __restrict__ out)      // [100000,1000]
{
    const int lane   = threadIdx.x & 31;
    const int wave   = threadIdx.x >> 5;
    const int m_tile = blockIdx.x * 8 + wave;
    if (m_tile >= M_TILES) return;       // uniform per wave -> EXEC stays all-1s

    const int sub  = lane & 15;
    const int half = lane >> 4;          // 0 or 1
    const int kb   = half * 2;           // K sub-offset inside each 4-wide block

    const float* Srow = S + (size_t)(m_tile * 16 + sub) * NCOEF;
    v2f a0 = { Srow[0 + kb], Srow[1 + kb] };
    v2f a1 = { Srow[4 + kb], Srow[5 + kb] };
    v2f a2 = { Srow[8 + kb], Srow[9 + kb] };

    const int nt0 = blockIdx.y * 8;
    for (int q = 0; q < 8; ++q) {
        int nt = nt0 + q;
        if (nt >= N_TILES) break;        // uniform per wave
        int col = nt * 16 + sub;

        v2f b0 = { Bmat[(0 + kb) * BP + col], Bmat[(1 + kb) * BP + col] };
        v2f b1 = { Bmat[(4 + kb) * BP + col], Bmat[(5 + kb) * BP + col] };
        v2f b2 = { Bmat[(8 + kb) * BP + col], Bmat[(9 + kb) * BP + col] };

        v8f c = {};
        c = __builtin_amdgcn_wmma_f32_16x16x4_f32(false, a0, false, b0,
                                                  (short)0, c, false, false);
        c = __builtin_amdgcn_wmma_f32_16x16x4_f32(false, a1, false, b1,
                                                  (short)0, c, false, false);
        c = __builtin_amdgcn_wmma_f32_16x16x4_f32(false, a2, false, b2,
                                                  (short)0, c, false, false);

        if (col < N_TIME) {              // only divergence: store guard (legal)
            float* o = out + (size_t)(m_tile * 16 + half * 8) * N_TIME + col;
            #pragma unroll
            for (int r = 0; r < 8; ++r)
                o[(size_t)r * N_TIME] = c[r];
        }
    }
}

// ---------------------------------------------------------------------------
extern "C" void kernel_launch(void* const* d_in, const int* in_sizes, int n_in,
                              void* d_out, int out_size, void* d_ws, size_t ws_size,
                              hipStream_t stream) {
    (void)in_sizes; (void)n_in; (void)out_size; (void)ws_size;

    const float* time_vector = (const float*)d_in[0];
    const float* coff        = (const float*)d_in[1];
    const float* trend       = (const float*)d_in[2];
    const float* amps        = (const float*)d_in[3];
    const float* phases      = (const float*)d_in[4];
    const int*   nbr_idx     = (const int*)  d_in[5];
    const float* nbr_w       = (const float*)d_in[6];
    const float* periods     = (const float*)d_in[7];
    float* out = (float*)d_out;

    float* S    = (float*)d_ws;                       // 100000*12 f32 = 4.8 MB
    float* Bmat = S + (size_t)N_STATIONS * NCOEF;     // 12*1008 f32

    coeff_kernel<<<(N_STATIONS + 255) / 256, 256, 0, stream>>>(
        amps, phases, nbr_idx, nbr_w, coff, trend, S);

    basis_kernel<<<(BP + 255) / 256, 256, 0, stream>>>(time_vector, periods, Bmat);

    dim3 grid((M_TILES + 7) / 8, (N_TILES + 7) / 8);  // 782 x 8 blocks, 8 waves each
    gemm_kernel<<<grid, 256, 0, stream>>>(S, Bmat, out);
}